// GraphBasedLSTMClassifier_4741643895064
// MI455X (gfx1250) — compile-verified
//
#include <hip/hip_runtime.h>
#include <cstdint>
#include <cmath>

// Problem dims (compile-time constants from the reference)
#define T_DIM 36
#define B_DIM 16
#define N_DIM 2048
#define F_DIM 64
#define H_DIM 256
#define E_DIM 32768
#define C1_DIM 16
#define K_TOP 1639          // ceil(0.8*2048)
#define G4H (4*H_DIM)       // 1024
#define TILE_C 4            // channel tile for fused GCN2 (2048*4*4B = 32KB LDS)
#define ROWS_PER_BLK 128    // GEMM1: 128 rows x 64 cols fp32 = 32KB LDS A-tile

static_assert(E_DIM % 256 == 0, "edge chunking");
static_assert(N_DIM % 16 == 0, "wmma tiling");

typedef float v2f __attribute__((ext_vector_type(2)));
typedef float v8f __attribute__((ext_vector_type(8)));
typedef unsigned int u32x4 __attribute__((ext_vector_type(4)));
typedef int i32x4 __attribute__((ext_vector_type(4)));
typedef int i32x8 __attribute__((ext_vector_type(8)));

#if defined(__gfx1250__) && __has_builtin(__builtin_amdgcn_tensor_load_to_lds)
#define USE_TDM 1
#else
#define USE_TDM 0
#endif

__device__ __forceinline__ float sigmoidf_(float x) { return 1.0f / (1.0f + expf(-x)); }

// ---------------------------------------------------------------- utilities
__global__ void fill_kernel(float* __restrict__ p, size_t n, float v) {
    size_t i = (size_t)blockIdx.x * blockDim.x + threadIdx.x;
    if (i < n) p[i] = v;
}

// ---------------------------------------------------------------- degrees
// deg1[t][n] = indegree(t,n) + 1  (mask = ones; shared across batch)
__global__ void deg1_scatter(const int* __restrict__ ei, float* __restrict__ deg1) {
    int t = blockIdx.y;
    int e = blockIdx.x * blockDim.x + threadIdx.x;
    const int* eb = ei + (size_t)t * 2 * E_DIM;
    int d = eb[E_DIM + e];
    atomicAdd(&deg1[(size_t)t * N_DIM + d], 1.0f);
}

// ---------------------------------------------------------------- GEMM1: h1pre = x @ W1
// TDM stages a 128x64 fp32 A super-tile into LDS (one tensor_load_to_lds per
// block, waited with s_wait_tensorcnt); 8 waves then run fp32 WMMA 16x16x4
// (16 k-steps) reading A fragments from LDS and W1 from global (L2-resident).
__global__ void gemm1_wmma(const float* __restrict__ x, const float* __restrict__ W1,
                           float* __restrict__ h1pre) {
    __shared__ __align__(16) float Atile[ROWS_PER_BLK * F_DIM];   // 32 KB

    const int wv   = threadIdx.x >> 5;        // wave in block: 0..7
    const int lane = threadIdx.x & 31;
    const int half = lane >> 4;               // 0: lanes 0-15, 1: lanes 16-31
    const int r    = lane & 15;
    const size_t blk_row = (size_t)blockIdx.x * ROWS_PER_BLK;

#if USE_TDM
    if (wv == 0) {
        // Tensor DMA descriptor (D#), 2-D tensor: 128 rows x 64 fp32, contiguous.
        unsigned long long gaddr = (unsigned long long)(const void*)(x + blk_row * F_DIM);
        unsigned int lds_addr = (unsigned int)(size_t)(&Atile[0]);   // low 32b = LDS offset
        u32x4 g0;
        g0.x = 1u;                                     // count=1, user mode
        g0.y = lds_addr;                               // lds_addr [63:32]
        g0.z = (unsigned int)(gaddr & 0xffffffffull);  // global_addr [95:64]
        g0.w = (unsigned int)((gaddr >> 32) & 0x01ffffffull) | (2u << 30); // addr hi | type=2
        i32x8 g1;
        g1[0] = 0x00020000;                 // workgroup_mask=0, data_size=2 (4B)
        g1[1] = (F_DIM & 0xffff) << 16;     // tensor_dim0 = 64 (bits 79:48)
        g1[2] = (ROWS_PER_BLK & 0xffff) << 16; // tensor_dim1 = 128 (bits 111:80)
        g1[3] = (F_DIM & 0xffff) << 16;     // tile_dim0 = 64 (bits 127:112)
        g1[4] = ROWS_PER_BLK;               // tile_dim1 = 128 (bits 143:128)
        g1[5] = F_DIM;                      // tensor_dim0_stride = 64 (bits 207:160)
        g1[6] = 0;
        g1[7] = 0;
        i32x4 gz4 = {0, 0, 0, 0};           // groups 2/3 unused (2-D tensor)
        i32x8 gz8 = {0, 0, 0, 0, 0, 0, 0, 0};
        __builtin_amdgcn_tensor_load_to_lds(g0, g1, gz4, gz4, gz8, 0);
        __builtin_amdgcn_s_wait_tensorcnt(0);
    }
#else
    // Fallback: coalesced cooperative float4 staging into LDS.
    {
        const float4* src = (const float4*)(x + blk_row * F_DIM);
        float4* dst = (float4*)Atile;
        for (int i = threadIdx.x; i < ROWS_PER_BLK * F_DIM / 4; i += blockDim.x)
            dst[i] = src[i];
    }
#endif
    __syncthreads();

    const float* Arow = Atile + (wv * 16 + r) * F_DIM;
    v8f acc = {};
#pragma unroll
    for (int k0 = 0; k0 < F_DIM; k0 += 4) {
        v2f a, b;
        // A 16x4 frag: lanes<16 hold K=k0,k0+1 ; lanes>=16 hold K=k0+2,k0+3 (row=r)
        a.x = Arow[k0 + 2 * half];
        a.y = Arow[k0 + 2 * half + 1];
        // B 4x16 frag: lanes<16 rows K=k0,k0+1 ; lanes>=16 rows K=k0+2,k0+3 (col=r)
        const float* bx = W1 + (size_t)(k0 + 2 * half) * C1_DIM + r;
        b.x = bx[0]; b.y = bx[C1_DIM];
        acc = __builtin_amdgcn_wmma_f32_16x16x4_f32(false, a, false, b, (short)0, acc,
                                                    false, false);
    }
    // C/D layout: VGPR i -> M = i (lanes 0-15) or i+8 (lanes 16-31); N = r
    const size_t row_base = blk_row + (size_t)wv * 16;
#pragma unroll
    for (int i = 0; i < 8; i++) {
        int m = i + 8 * half;
        h1pre[(row_base + m) * C1_DIM + r] = acc[i];
    }
}

// ---------------------------------------------------------------- GCN1 scatter: agg1 += h1pre[src]*norm
__global__ void scatter1_kernel(const int* __restrict__ ei, const float* __restrict__ deg1,
                                const float* __restrict__ h1pre, float* __restrict__ agg1) {
    int tb = blockIdx.y;
    int t  = tb / B_DIM;
    int e  = blockIdx.x * blockDim.x + threadIdx.x;
    const int* eb = ei + (size_t)t * 2 * E_DIM;
    int s = eb[e], d = eb[E_DIM + e];
    float norm = rsqrtf(deg1[(size_t)t * N_DIM + s] * deg1[(size_t)t * N_DIM + d]);
    const float* hs = h1pre + ((size_t)tb * N_DIM + s) * C1_DIM;
    float* ad = agg1 + ((size_t)tb * N_DIM + d) * C1_DIM;
#pragma unroll
    for (int c = 0; c < C1_DIM; c++) atomicAdd(&ad[c], hs[c] * norm);
}

// h1 = relu(agg1 + h1pre/deg + b1)  (in place into agg1);  spre = h1 @ sagW
__global__ void fin1_kernel(const float* __restrict__ deg1, const float* __restrict__ h1pre,
                            const float* __restrict__ b1, const float* __restrict__ sagW,
                            float* __restrict__ h1io, float* __restrict__ spre) {
    size_t gid = (size_t)blockIdx.x * blockDim.x + threadIdx.x;
    if (gid >= (size_t)T_DIM * B_DIM * N_DIM) return;
    size_t tb = gid / N_DIM, n = gid % N_DIM, t = tb / B_DIM;
    float invd = 1.0f / deg1[t * N_DIM + n];
    const float* hp = h1pre + gid * C1_DIM;
    float* ho = h1io + gid * C1_DIM;
    float s = 0.f;
#pragma unroll
    for (int c = 0; c < C1_DIM; c++) {
        float v = ho[c] + hp[c] * invd + b1[c];
        v = fmaxf(v, 0.f);
        ho[c] = v;
        s += v * sagW[c];
    }
    spre[gid] = s;
}

// SAG score GCN (1 channel)
__global__ void scatterS_kernel(const int* __restrict__ ei, const float* __restrict__ deg1,
                                const float* __restrict__ spre, float* __restrict__ sagg) {
    int tb = blockIdx.y;
    int t  = tb / B_DIM;
    int e  = blockIdx.x * blockDim.x + threadIdx.x;
    const int* eb = ei + (size_t)t * 2 * E_DIM;
    int s = eb[e], d = eb[E_DIM + e];
    float norm = rsqrtf(deg1[(size_t)t * N_DIM + s] * deg1[(size_t)t * N_DIM + d]);
    atomicAdd(&sagg[(size_t)tb * N_DIM + d], spre[(size_t)tb * N_DIM + s] * norm);
}

__global__ void finS_kernel(const float* __restrict__ deg1, const float* __restrict__ spre,
                            const float* __restrict__ sagb, float* __restrict__ sagg) {
    size_t gid = (size_t)blockIdx.x * blockDim.x + threadIdx.x;
    if (gid >= (size_t)T_DIM * B_DIM * N_DIM) return;
    size_t tb = gid / N_DIM, n = gid % N_DIM, t = tb / B_DIM;
    float invd = 1.0f / deg1[t * N_DIM + n];
    sagg[gid] = tanhf(sagg[gid] + spre[gid] * invd + sagb[0]);
}

// ---------------------------------------------------------------- top-K per graph (bitonic in LDS)
__global__ void topk_kernel(const float* __restrict__ score, const float* __restrict__ h1,
                            float* __restrict__ xp, int* __restrict__ newidx) {
    __shared__ float key[N_DIM];
    __shared__ int   idx[N_DIM];
    int tb = blockIdx.x;
    const float* sc = score + (size_t)tb * N_DIM;
    for (int i = threadIdx.x; i < N_DIM; i += blockDim.x) { key[i] = sc[i]; idx[i] = i; }
    __syncthreads();
    // Bitonic sort, descending by key, ties -> lower original index first (top_k semantics)
    for (int k = 2; k <= N_DIM; k <<= 1) {
        for (int j = k >> 1; j > 0; j >>= 1) {
            for (int i = threadIdx.x; i < N_DIM; i += blockDim.x) {
                int ixj = i ^ j;
                if (ixj > i) {
                    float k1 = key[i], k2 = key[ixj];
                    int   i1 = idx[i], i2 = idx[ixj];
                    bool firstGreater = (k1 > k2) || (k1 == k2 && i1 < i2);
                    bool desc = ((i & k) == 0);
                    bool doswap = desc ? (!firstGreater) : firstGreater;
                    if (doswap) { key[i] = k2; key[ixj] = k1; idx[i] = i2; idx[ixj] = i1; }
                }
            }
            __syncthreads();
        }
    }
    int* ni = newidx + (size_t)tb * N_DIM;
    for (int i = threadIdx.x; i < N_DIM; i += blockDim.x) ni[i] = -1;
    __syncthreads();
    const float* h1p = h1 + (size_t)tb * N_DIM * C1_DIM;
    float* xpp = xp + (size_t)tb * N_DIM * C1_DIM;
    for (int r = threadIdx.x; r < N_DIM; r += blockDim.x) {
        if (r < K_TOP) {
            int n = idx[r];
            ni[n] = r;
            float v = key[r];
#pragma unroll
            for (int c = 0; c < C1_DIM; c++) xpp[(size_t)r * C1_DIM + c] = h1p[(size_t)n * C1_DIM + c] * v;
        } else {
#pragma unroll
            for (int c = 0; c < C1_DIM; c++) xpp[(size_t)r * C1_DIM + c] = 0.f;
        }
    }
}

__global__ void deg2_scatter(const int* __restrict__ ei, const int* __restrict__ newidx,
                             float* __restrict__ deg2) {
    int tb = blockIdx.y;
    int t  = tb / B_DIM;
    int e  = blockIdx.x * blockDim.x + threadIdx.x;
    const int* eb = ei + (size_t)t * 2 * E_DIM;
    const int* ni = newidx + (size_t)tb * N_DIM;
    int ns = ni[eb[e]], nd = ni[eb[E_DIM + e]];
    if (ns >= 0 && nd >= 0) atomicAdd(&deg2[(size_t)tb * N_DIM + nd], 1.0f);
}

// ---------------------------------------------------------------- fused GCN2 + mean (per graph, per 4-channel tile)
// Recomputes h2pre(row,c) = dot(xp[row][0:16], W2[:,c]) on the fly; aggregates in LDS.
__global__ void gcn2_kernel(const int* __restrict__ ei, const int* __restrict__ newidx,
                            const float* __restrict__ deg2, const float* __restrict__ xp,
                            const float* __restrict__ W2, const float* __restrict__ b2,
                            float* __restrict__ emb) {
    __shared__ float agg[N_DIM * TILE_C];   // 32 KB
    __shared__ float w2s[C1_DIM * TILE_C];
    __shared__ float csum[TILE_C];
    int tb = blockIdx.x;
    int t  = tb / B_DIM;
    int c0 = blockIdx.y * TILE_C;
    for (int i = threadIdx.x; i < N_DIM * TILE_C; i += blockDim.x) agg[i] = 0.f;
    if (threadIdx.x < C1_DIM * TILE_C) {
        int k = threadIdx.x / TILE_C, c = threadIdx.x % TILE_C;
        w2s[threadIdx.x] = W2[(size_t)k * H_DIM + c0 + c];
    }
    if (threadIdx.x < TILE_C) csum[threadIdx.x] = 0.f;
    __syncthreads();

    const int* esrc = ei + (size_t)t * 2 * E_DIM;
    const int* edst = esrc + E_DIM;
    const int* ni   = newidx + (size_t)tb * N_DIM;
    const float* dg  = deg2 + (size_t)tb * N_DIM;
    const float* xpb = xp + (size_t)tb * N_DIM * C1_DIM;

    for (int e = threadIdx.x; e < E_DIM; e += blockDim.x) {
        int ns = ni[esrc[e]], nd = ni[edst[e]];
        if (ns >= 0 && nd >= 0) {
            float norm = rsqrtf(dg[ns] * dg[nd]);
            const float* xr = xpb + (size_t)ns * C1_DIM;
#pragma unroll
            for (int c = 0; c < TILE_C; c++) {
                float hh = 0.f;
#pragma unroll
                for (int k = 0; k < C1_DIM; k++) hh += xr[k] * w2s[k * TILE_C + c];
                atomicAdd(&agg[nd * TILE_C + c], hh * norm);   // ds_add_f32
            }
        }
    }
    __syncthreads();

    float ls[TILE_C] = {0.f, 0.f, 0.f, 0.f};
    for (int r = threadIdx.x; r < K_TOP; r += blockDim.x) {
        const float* xr = xpb + (size_t)r * C1_DIM;
        float invd = 1.0f / dg[r];
#pragma unroll
        for (int c = 0; c < TILE_C; c++) {
            float hh = 0.f;
#pragma unroll
            for (int k = 0; k < C1_DIM; k++) hh += xr[k] * w2s[k * TILE_C + c];
            float v = agg[r * TILE_C + c] + hh * invd + b2[c0 + c];
            ls[c] += fmaxf(v, 0.f);
        }
    }
#pragma unroll
    for (int c = 0; c < TILE_C; c++) atomicAdd(&csum[c], ls[c]);
    __syncthreads();
    if (threadIdx.x < TILE_C)
        emb[(size_t)tb * H_DIM + c0 + threadIdx.x] = csum[threadIdx.x] / (float)K_TOP;
}

// ---------------------------------------------------------------- LSTM gates GEMM (fp32 WMMA)
// gates(16 x 1024) = xt(16x256) @ W_ih^T + h(16x256) @ W_hh^T + biases.
// One wave per 16-wide N tile; M=16 = batch; K loop of 4.
__global__ void lstm_gemm_wmma(const float* __restrict__ xt, const float* __restrict__ hbuf,
                               const float* __restrict__ W_ih, const float* __restrict__ W_hh,
                               const float* __restrict__ b_ih, const float* __restrict__ b_hh,
                               float* __restrict__ gates) {
    int lane = threadIdx.x & 31;
    int half = lane >> 4, r = lane & 15;
    int n = blockIdx.x * 16 + r;                 // output gate column for this lane
    v8f acc = {};
#pragma unroll 4
    for (int k0 = 0; k0 < H_DIM; k0 += 4) {
        v2f a, b;
        a.x = xt[(size_t)r * H_DIM + k0 + 2 * half];
        a.y = xt[(size_t)r * H_DIM + k0 + 2 * half + 1];
        b.x = W_ih[(size_t)n * H_DIM + k0 + 2 * half];      // B[k][n] = W_ih[n][k]
        b.y = W_ih[(size_t)n * H_DIM + k0 + 2 * half + 1];
        acc = __builtin_amdgcn_wmma_f32_16x16x4_f32(false, a, false, b, (short)0, acc,
                                                    false, false);
    }
#pragma unroll 4
    for (int k0 = 0; k0 < H_DIM; k0 += 4) {
        v2f a, b;
        a.x = hbuf[(size_t)r * H_DIM + k0 + 2 * half];
        a.y = hbuf[(size_t)r * H_DIM + k0 + 2 * half + 1];
        b.x = W_hh[(size_t)n * H_DIM + k0 + 2 * half];
        b.y = W_hh[(size_t)n * H_DIM + k0 + 2 * half + 1];
        acc = __builtin_amdgcn_wmma_f32_16x16x4_f32(false, a, false, b, (short)0, acc,
                                                    false, false);
    }
    float bias = b_ih[n] + b_hh[n];
#pragma unroll
    for (int i = 0; i < 8; i++) {
        int m = i + 8 * half;
        gates[(size_t)m * G4H + n] = acc[i] + bias;
    }
}

__global__ void lstm_update(const float* __restrict__ gates, float* __restrict__ cbuf,
                            float* __restrict__ hbuf) {
    int m = blockIdx.x, j = threadIdx.x;
    const float* g = gates + (size_t)m * G4H;
    float ig = sigmoidf_(g[j]);
    float fg = sigmoidf_(g[H_DIM + j]);
    float gg = tanhf(g[2 * H_DIM + j]);
    float og = sigmoidf_(g[3 * H_DIM + j]);
    float c  = fg * cbuf[(size_t)m * H_DIM + j] + ig * gg;
    cbuf[(size_t)m * H_DIM + j] = c;
    hbuf[(size_t)m * H_DIM + j] = og * tanhf(c);
}

__global__ void out_kernel(const float* __restrict__ hbuf, const float* __restrict__ outW,
                           const float* __restrict__ outb, float* __restrict__ out) {
    int b = threadIdx.x;
    if (b < B_DIM) {
        float s = 0.f;
        for (int k = 0; k < H_DIM; k++) s += hbuf[(size_t)b * H_DIM + k] * outW[k];
        out[b] = sigmoidf_(s + outb[0]);
    }
}

// ---------------------------------------------------------------- launch
extern "C" void kernel_launch(void* const* d_in, const int* in_sizes, int n_in,
                              void* d_out, int out_size, void* d_ws, size_t ws_size,
                              hipStream_t stream) {
    const float* x    = (const float*)d_in[0];
    const int*   ei   = (const int*)d_in[1];   // (T, 2, E) int32
    const float* W1   = (const float*)d_in[2];
    const float* b1   = (const float*)d_in[3];
    const float* sagW = (const float*)d_in[4];
    const float* sagb = (const float*)d_in[5];
    const float* W2   = (const float*)d_in[6];
    const float* b2   = (const float*)d_in[7];
    const float* W_ih = (const float*)d_in[8];
    const float* W_hh = (const float*)d_in[9];
    const float* b_ih = (const float*)d_in[10];
    const float* b_hh = (const float*)d_in[11];
    const float* outW = (const float*)d_in[12];
    const float* outb = (const float*)d_in[13];
    float* out = (float*)d_out;
    (void)in_sizes; (void)n_in; (void)out_size; (void)ws_size;

    const size_t TB = (size_t)T_DIM * B_DIM;
    float* ws = (float*)d_ws;
    size_t off = 0;
    float* deg1  = ws + off; off += (size_t)T_DIM * N_DIM;
    float* h1pre = ws + off; off += TB * N_DIM * C1_DIM;   // later reused as xp
    float* agg1  = ws + off; off += TB * N_DIM * C1_DIM;   // becomes h1 in place
    float* spre  = ws + off; off += TB * N_DIM;
    float* sagg  = ws + off; off += TB * N_DIM;            // becomes score in place
    int*   newidx = (int*)(ws + off); off += TB * N_DIM;
    float* deg2  = ws + off; off += TB * N_DIM;
    float* emb   = ws + off; off += TB * H_DIM;
    float* gates = ws + off; off += (size_t)B_DIM * G4H;
    float* hbuf  = ws + off; off += (size_t)B_DIM * H_DIM;
    float* cbuf  = ws + off; off += (size_t)B_DIM * H_DIM;

    float* xp    = h1pre;   // h1pre dead after fin1
    float* h1    = agg1;
    float* score = sagg;

    auto nblk = [](size_t n, int bs) { return (unsigned)((n + (size_t)bs - 1) / bs); };

    // GCN1 degrees
    fill_kernel<<<nblk((size_t)T_DIM * N_DIM, 256), 256, 0, stream>>>(deg1, (size_t)T_DIM * N_DIM, 1.0f);
    deg1_scatter<<<dim3(E_DIM / 256, T_DIM), 256, 0, stream>>>(ei, deg1);

    // GCN1: GEMM (TDM-staged WMMA) + scatter + finalize (+ SAG projection)
    fill_kernel<<<nblk(TB * N_DIM * C1_DIM, 256), 256, 0, stream>>>(agg1, TB * N_DIM * C1_DIM, 0.0f);
    gemm1_wmma<<<(unsigned)(TB * N_DIM / ROWS_PER_BLK), 256, 0, stream>>>(x, W1, h1pre);
    scatter1_kernel<<<dim3(E_DIM / 256, (unsigned)TB), 256, 0, stream>>>(ei, deg1, h1pre, agg1);
    fin1_kernel<<<nblk(TB * N_DIM, 256), 256, 0, stream>>>(deg1, h1pre, b1, sagW, agg1, spre);

    // SAG score GCN
    fill_kernel<<<nblk(TB * N_DIM, 256), 256, 0, stream>>>(sagg, TB * N_DIM, 0.0f);
    scatterS_kernel<<<dim3(E_DIM / 256, (unsigned)TB), 256, 0, stream>>>(ei, deg1, spre, sagg);
    finS_kernel<<<nblk(TB * N_DIM, 256), 256, 0, stream>>>(deg1, spre, sagb, sagg);

    // top-K pooling
    topk_kernel<<<(unsigned)TB, 256, 0, stream>>>(score, h1, xp, newidx);
    fill_kernel<<<nblk(TB * N_DIM, 256), 256, 0, stream>>>(deg2, TB * N_DIM, 1.0f);
    deg2_scatter<<<dim3(E_DIM / 256, (unsigned)TB), 256, 0, stream>>>(ei, newidx, deg2);

    // fused GCN2 + mean -> emb
    gcn2_kernel<<<dim3((unsigned)TB, H_DIM / TILE_C), 256, 0, stream>>>(ei, newidx, deg2, xp, W2, b2, emb);

    // LSTM over T steps (hbuf & cbuf are contiguous -> one fill)
    fill_kernel<<<nblk((size_t)B_DIM * H_DIM * 2, 256), 256, 0, stream>>>(hbuf, (size_t)B_DIM * H_DIM * 2, 0.0f);
    for (int t = 0; t < T_DIM; t++) {
        lstm_gemm_wmma<<<G4H / 16, 32, 0, stream>>>(emb + (size_t)t * B_DIM * H_DIM, hbuf,
                                                    W_ih, W_hh, b_ih, b_hh, gates);
        lstm_update<<<B_DIM, H_DIM, 0, stream>>>(gates, cbuf, hbuf);
    }
    out_kernel<<<1, 64, 0, stream>>>(hbuf, outW, outb, out);
}